// SetPool_42734924595834
// MI455X (gfx1250) — compile-verified
//
#include <hip/hip_runtime.h>
#include <stdint.h>

typedef float v2f __attribute__((ext_vector_type(2)));
typedef float v8f __attribute__((ext_vector_type(8)));

#define B_   16
#define K_   21
#define F_   64
#define N_   65536   // H*W = 256*256
// d_ws layout:
//   [0, B_*F_*K_*4)            float acc[B][F][K]   (84 KB)
//   [131072, 131072 + B_*N_)   uint8 labels[B][N]   (1 MB)
#define WS_LABEL_OFF (128 * 1024)

// ---------------- Pass 0: zero the accumulator ----------------
__global__ void zero_acc_kernel(float* __restrict__ acc) {
    int i = blockIdx.x * blockDim.x + threadIdx.x;
    if (i < B_ * F_ * K_) acc[i] = 0.0f;
}

// ---------------- Pass 1: per-pixel argmax over K=21 ----------------
// grid covers B*N threads; coalesced strided reads of outputs[b,k,:].
__global__ void __launch_bounds__(256) argmax_kernel(const float* __restrict__ outputs,
                                                     uint8_t* __restrict__ labels) {
    int gid = blockIdx.x * blockDim.x + threadIdx.x;   // 0 .. B*N-1
    int b = gid >> 16;            // N_ = 2^16
    int n = gid & (N_ - 1);
    const float* p = outputs + ((size_t)b * K_) * N_ + n;
    float best = p[0];
    int bi = 0;
#pragma unroll
    for (int k = 1; k < K_; ++k) {
        float v = p[(size_t)k * N_];
        if (v > best) { best = v; bi = k; }   // strict > : first max wins, matches argmax
    }
    labels[gid] = (uint8_t)bi;
}

// ---------------- Pass 2: segment-sum as one-hot WMMA bmm ----------------
// Each wave owns one batch b and a 512-pixel stripe; accumulates the FULL
// [F=64 x Kpad=32] product in 8 WMMA f32 accumulator tiles held in VGPRs.
// A tile (16x4 f32):  lane m=lane&15 holds feats[f, n0+j0 .. n0+j0+1] (j0 = 0 or 2 per lane-half)
// B tile (4x16 f32):  lane col=lane&15 holds onehot(label[n0+j0(+1)] == class) — built in registers
__global__ void __launch_bounds__(256) segsum_wmma_kernel(const float* __restrict__ feats,
                                                          const uint8_t* __restrict__ labels,
                                                          float* __restrict__ acc) {
    const int lane   = threadIdx.x & 31;
    const int wave   = threadIdx.x >> 5;
    const int b      = blockIdx.x >> 4;                    // 16 blocks per batch
    const int stripe = ((blockIdx.x & 15) << 3) | wave;    // 0..127
    const int nbase  = stripe << 9;                        // * 512 pixels

    const int m    = lane & 15;        // A row / B,D column
    const int half = lane >> 4;        // lane-half selects K rows {0,1} vs {2,3}
    const int j0   = half << 1;

    v8f accT[4][2];
#pragma unroll
    for (int mt = 0; mt < 4; ++mt)
#pragma unroll
        for (int nt = 0; nt < 2; ++nt)
            accT[mt][nt] = (v8f){0.f, 0.f, 0.f, 0.f, 0.f, 0.f, 0.f, 0.f};

    const uint8_t* lab = labels + ((size_t)b << 16);
    const float*   fb  = feats + (((size_t)(b * F_)) << 16);

    for (int it = 0; it < 128; ++it) {
        const int n0 = nbase + (it << 2);          // 4 pixels per iteration
        const uint32_t l4  = *(const uint32_t*)(lab + n0);   // 4 packed labels
        const uint32_t lk0 = (l4 >> (8 * j0)) & 0xFFu;       // label(n0 + j0)
        const uint32_t lk1 = (l4 >> (8 * j0 + 8)) & 0xFFu;   // label(n0 + j0 + 1)

        v2f Bt[2];
#pragma unroll
        for (int nt = 0; nt < 2; ++nt) {
            const uint32_t cls = (uint32_t)(nt * 16 + m);
            Bt[nt].x = (lk0 == cls) ? 1.0f : 0.0f;
            Bt[nt].y = (lk1 == cls) ? 1.0f : 0.0f;
        }

#pragma unroll
        for (int mt = 0; mt < 4; ++mt) {
            const float* ap = fb + (((size_t)(mt * 16 + m)) << 16) + (n0 + j0);
            v2f At = *(const v2f*)ap;              // 8B-aligned global_load_b64
#pragma unroll
            for (int nt = 0; nt < 2; ++nt) {
                accT[mt][nt] = __builtin_amdgcn_wmma_f32_16x16x4_f32(
                    /*neg_a=*/false, At, /*neg_b=*/false, Bt[nt],
                    /*c_mod=*/(short)0, accT[mt][nt],
                    /*reuse_a=*/false, /*reuse_b=*/false);
            }
        }
    }

    // Flush: D tile layout — VGPR v holds row (v + 8*half), lane&15 is the column.
#pragma unroll
    for (int mt = 0; mt < 4; ++mt) {
#pragma unroll
        for (int nt = 0; nt < 2; ++nt) {
#pragma unroll
            for (int v = 0; v < 8; ++v) {
                const int f = mt * 16 + v + half * 8;
                const int k = nt * 16 + m;
                if (k < K_)
                    atomicAdd(&acc[((size_t)b * F_ + f) * K_ + k], accT[mt][nt][v]);
            }
        }
    }
}

// ---------------- Pass 3: BatchNorm over (B,K) per channel F ----------------
__global__ void __launch_bounds__(128) bn_kernel(const float* __restrict__ acc,
                                                 const float* __restrict__ gamma,
                                                 const float* __restrict__ beta,
                                                 float* __restrict__ out) {
    const int f = blockIdx.x;
    const int t = threadIdx.x;
    __shared__ float xs[B_ * K_];   // 336 values of this channel
    __shared__ float red[128];

    for (int i = t; i < B_ * K_; i += 128) {
        const int b = i / K_, k = i % K_;
        xs[i] = acc[((size_t)b * F_ + f) * K_ + k];
    }
    __syncthreads();

    float s = 0.f;
    for (int i = t; i < B_ * K_; i += 128) s += xs[i];
    red[t] = s; __syncthreads();
    for (int off = 64; off > 0; off >>= 1) {
        if (t < off) red[t] += red[t + off];
        __syncthreads();
    }
    const float mean = red[0] * (1.0f / (B_ * K_));
    __syncthreads();

    float sq = 0.f;
    for (int i = t; i < B_ * K_; i += 128) { float d = xs[i] - mean; sq += d * d; }
    red[t] = sq; __syncthreads();
    for (int off = 64; off > 0; off >>= 1) {
        if (t < off) red[t] += red[t + off];
        __syncthreads();
    }
    const float var   = red[0] * (1.0f / (B_ * K_));   // biased variance, as reference
    const float scale = rsqrtf(var + 1e-5f) * gamma[f];
    const float shift = beta[f];

    for (int i = t; i < B_ * K_; i += 128) {
        const int b = i / K_, k = i % K_;
        out[((size_t)b * F_ + f) * K_ + k] = (xs[i] - mean) * scale + shift;
    }
}

extern "C" void kernel_launch(void* const* d_in, const int* in_sizes, int n_in,
                              void* d_out, int out_size, void* d_ws, size_t ws_size,
                              hipStream_t stream) {
    const float* outputs = (const float*)d_in[0];   // [B,K,H,W] f32
    const float* feats   = (const float*)d_in[1];   // [B,F,H,W] f32
    const float* gamma   = (const float*)d_in[2];   // [F] f32
    const float* beta    = (const float*)d_in[3];   // [F] f32
    float*   out    = (float*)d_out;                // [B,F,K] f32
    float*   acc    = (float*)d_ws;
    uint8_t* labels = (uint8_t*)d_ws + WS_LABEL_OFF;

    zero_acc_kernel<<<(B_ * F_ * K_ + 255) / 256, 256, 0, stream>>>(acc);
    argmax_kernel<<<(B_ * N_) / 256, 256, 0, stream>>>(outputs, labels);
    segsum_wmma_kernel<<<B_ * 16, 256, 0, stream>>>(feats, labels, acc);
    bn_kernel<<<F_, 128, 0, stream>>>(acc, gamma, beta, out);
}